// ZBL_5068061409422
// MI455X (gfx1250) — compile-verified
//
#include <hip/hip_runtime.h>
#include <hip/hip_bf16.h>
#include <math.h>

#define TILE   1024
#define BLOCK  256

// Low 32 bits of a flat shared-memory pointer are the LDS byte offset
// (aperture only occupies addr[63:32] on CDNA5; flat->LDS uses addr[31:0]).
__device__ __forceinline__ uint32_t lds_off32(const void* p) {
  return (uint32_t)(uintptr_t)p;
}

// CDNA5 async copy: 16 bytes per lane, global (saddr base + 32-bit voffset) -> LDS.
// Tracked by ASYNCcnt, drained with s_wait_asynccnt.
__device__ __forceinline__ void async_cp_b128(uint32_t lds, const void* gbase,
                                              uint32_t byte_off) {
  asm volatile("global_load_async_to_lds_b128 %0, %1, %2"
               :
               : "v"(lds), "v"(byte_off), "s"(gbase)
               : "memory");
}

// Per-edge ZBL evaluation using the precomputed 16-entry type-pair table.
// Table layout per pair (16 floats, 64B record -> shift-only addressing):
//  [0..3] = -da_k * log2(e)   (so phi uses raw v_exp_f32)
//  [4] = 0.5*factor  [5] = rc  [6] = A/6  [7] = B/8  [8] = C/2  [9..15] pad
// max_rc early-out skips the two random L2 gathers + table reads + exps for
// the ~28% of edges that cannot contribute (masked lanes make no mem requests).
__device__ __forceinline__ void zbl_process(float r, int fa, int sa, float max_rc,
                                            const int* __restrict__ atype,
                                            const float* __restrict__ tab,
                                            float* __restrict__ out) {
  if (r <= max_rc) {
    const int ti = atype[fa];
    const int tj = atype[sa];
    const float* T = tab + (((ti << 2) | tj) << 4);
    const float4 md = *(const float4*)(T);      // -da_k*log2e
    const float4 cf = *(const float4*)(T + 4);  // hf, rc, A6, B8
    const float  C2 = T[8];
    if (r <= cf.y) {
      float p = 0.02817f * __builtin_amdgcn_exp2f(r * md.x)
              + 0.28022f * __builtin_amdgcn_exp2f(r * md.y)
              + 0.50986f * __builtin_amdgcn_exp2f(r * md.z)
              + 0.18175f * __builtin_amdgcn_exp2f(r * md.w);
      float e = cf.x * __builtin_amdgcn_rcpf(r) * p
              + ((cf.z + cf.w * r) * r) * r * r + C2;
#if defined(__AMDGCN__)
      unsafeAtomicAdd(&out[fa], e);   // native global_atomic_add_f32
#else
      atomicAdd(&out[fa], e);
#endif
    }
  }
}

__global__ __launch_bounds__(BLOCK) void zbl_edge_kernel(
    const float* __restrict__ rij,
    const float* __restrict__ rcov,
    const float* __restrict__ znum,
    const int*   __restrict__ first_atom,
    const int*   __restrict__ second_atom,
    const int*   __restrict__ atype,
    float*       __restrict__ out,
    int E, int ntiles) {
  __shared__ __align__(16) float s_r[2][TILE];
  __shared__ __align__(16) int   s_f[2][TILE];
  __shared__ __align__(16) int   s_s[2][TILE];
  __shared__ __align__(16) float s_tab[16 * 16];

  const int tid = threadIdx.x;

  // ---- Build the 16-entry type-pair constant table (f32, mirrors reference order) ----
  if (tid < 16) {
    const int ti = tid >> 2, tj = tid & 3;
    const float c0 = 0.02817f, c1 = 0.28022f, c2 = 0.50986f, c3 = 0.18175f;
    const float d0 = 0.20162f, d1 = 0.4029f,  d2 = 0.94229f, d3 = 3.1998f;
    const float zi = znum[ti], zj = znum[tj];
    const float rc = rcov[ti] + rcov[tj];
    const float a  = 0.4685f / (powf(zi, 0.23f) + powf(zj, 0.23f));
    const float ai = 1.0f / a;
    const float da0 = d0 * ai, da1 = d1 * ai, da2 = d2 * ai, da3 = d3 * ai;
    const float e0 = expf(-rc * da0), e1 = expf(-rc * da1);
    const float e2 = expf(-rc * da2), e3 = expf(-rc * da3);
    const float phi   = c0 * e0 + c1 * e1 + c2 * e2 + c3 * e3;
    const float dphi  = -(c0 * da0 * e0 + c1 * da1 * e1 + c2 * da2 * e2 + c3 * da3 * e3);
    const float d2phi = c0 * da0 * da0 * e0 + c1 * da1 * da1 * e1
                      + c2 * da2 * da2 * e2 + c3 * da3 * da3 * e3;
    const float fac = 14.399645478425668f * zi * zj;
    const float ri  = 1.0f / rc;
    const float ec   = fac * ri * phi;
    const float dec  = fac * ri * (-phi * ri + dphi);
    const float d2ec = fac * ri * (d2phi - 2.0f * ri * dphi + 2.0f * phi * ri * ri);
    const float A = (-3.0f * dec + rc * d2ec) * ri * ri;
    const float B = ( 2.0f * dec - rc * d2ec) * ri * ri * ri;
    const float C = -ec + 0.5f * rc * dec - rc * rc * d2ec * (1.0f / 12.0f);
    float* T = &s_tab[tid * 16];
    const float L2E = 1.4426950408889634f;
    T[0] = -da0 * L2E; T[1] = -da1 * L2E; T[2] = -da2 * L2E; T[3] = -da3 * L2E;
    T[4] = 0.5f * fac; T[5] = rc;
    T[6] = A * (1.0f / 6.0f); T[7] = B * 0.125f; T[8] = 0.5f * C;
    T[9] = 0.f; T[10] = 0.f; T[11] = 0.f;
    T[12] = 0.f; T[13] = 0.f; T[14] = 0.f; T[15] = 0.f;
  }
  __syncthreads();
  if (tid == 0) {  // reduce max cutoff for the cheap early-out
    float m = 0.0f;
    #pragma unroll
    for (int p = 0; p < 16; ++p) m = fmaxf(m, s_tab[p * 16 + 5]);
    s_tab[15] = m;
  }
  __syncthreads();
  const float max_rc = s_tab[15];

  // ---- Double-buffered async-to-LDS pipeline over 1024-edge tiles ----
  // Each thread stages & consumes its own 16B slice of each array, so no
  // per-tile barriers are needed; ASYNCcnt ordering is per-wave.
  int t = blockIdx.x;
  if (t < ntiles) {
    const uint32_t g = ((uint32_t)t * TILE + (uint32_t)tid * 4u) * 4u;
    async_cp_b128(lds_off32(&s_r[0][tid * 4]), rij,         g);
    async_cp_b128(lds_off32(&s_f[0][tid * 4]), first_atom,  g);
    async_cp_b128(lds_off32(&s_s[0][tid * 4]), second_atom, g);
  }
  int buf = 0;
  for (; t < ntiles; t += (int)gridDim.x, buf ^= 1) {
    const int tn = t + (int)gridDim.x;
    if (tn < ntiles) {
      // Prior tile's ds_loads fully drained before the async engine may
      // overwrite that buffer.
      asm volatile("s_wait_dscnt 0x0" ::: "memory");
      const uint32_t g = ((uint32_t)tn * TILE + (uint32_t)tid * 4u) * 4u;
      async_cp_b128(lds_off32(&s_r[buf ^ 1][tid * 4]), rij,         g);
      async_cp_b128(lds_off32(&s_f[buf ^ 1][tid * 4]), first_atom,  g);
      async_cp_b128(lds_off32(&s_s[buf ^ 1][tid * 4]), second_atom, g);
      asm volatile("s_wait_asynccnt 0x3" ::: "memory");  // current tile landed
    } else {
      asm volatile("s_wait_asynccnt 0x0" ::: "memory");
    }
    const float4 r4 = *(const float4*)&s_r[buf][tid * 4];
    const int4   f4 = *(const int4*)  &s_f[buf][tid * 4];
    const int4   s4 = *(const int4*)  &s_s[buf][tid * 4];
    zbl_process(r4.x, f4.x, s4.x, max_rc, atype, s_tab, out);
    zbl_process(r4.y, f4.y, s4.y, max_rc, atype, s_tab, out);
    zbl_process(r4.z, f4.z, s4.z, max_rc, atype, s_tab, out);
    zbl_process(r4.w, f4.w, s4.w, max_rc, atype, s_tab, out);
  }

  // ---- Tail (E not covered by full tiles): direct global loads ----
  for (int i = ntiles * TILE + (int)blockIdx.x * BLOCK + tid; i < E;
       i += (int)gridDim.x * BLOCK) {
    zbl_process(rij[i], first_atom[i], second_atom[i], max_rc, atype, s_tab, out);
  }
}

extern "C" void kernel_launch(void* const* d_in, const int* in_sizes, int n_in,
                              void* d_out, int out_size, void* d_ws, size_t ws_size,
                              hipStream_t stream) {
  const float* rij         = (const float*)d_in[0];
  const float* rcov        = (const float*)d_in[1];
  const float* znum        = (const float*)d_in[2];
  const int*   first_atom  = (const int*)d_in[3];
  const int*   second_atom = (const int*)d_in[4];
  const int*   atype       = (const int*)d_in[5];
  float*       out         = (float*)d_out;

  const int E       = in_sizes[0];
  const int n_atoms = out_size;
  const int ntiles  = E / TILE;

  // Output accumulator must start at zero (harness poisons it).
  hipMemsetAsync(out, 0, (size_t)n_atoms * sizeof(float), stream);

  int blocks = ntiles < 1248 ? ntiles : 1248;  // ~5 tiles/block at E=6.4M
  if (blocks < 1) blocks = 1;
  zbl_edge_kernel<<<blocks, BLOCK, 0, stream>>>(
      rij, rcov, znum, first_atom, second_atom, atype, out, E, ntiles);
}